// SegmentationCueBuilder_39762807227065
// MI455X (gfx1250) — compile-verified
//
#include <hip/hip_runtime.h>
#include <hip/hip_bf16.h>
#include <stdint.h>

// ---------------- problem constants ----------------
#define BATCH 64
#define NP 1024          // patches (32x32)
#define HID 1024
#define NC 150           // classes
#define NSLOT 11         // selected classes (NUM_SLOTS)
#define NTOK 16          // MAX_TOKENS
#define PW 32

#define SPLIT 4          // column-splits per batch for the GEMM (occupancy)
#define TPW 2            // 16x16 tiles per wave

typedef float v2f __attribute__((ext_vector_type(2)));
typedef float v8f __attribute__((ext_vector_type(8)));

// ---------------- reductions ----------------
__device__ __forceinline__ float blockReduceSum(float v, float* red) {
    int tid = threadIdx.x;
    red[tid] = v; __syncthreads();
    for (int s = 128; s > 0; s >>= 1) {
        if (tid < s) red[tid] += red[tid + s];
        __syncthreads();
    }
    float r = red[0]; __syncthreads();
    return r;
}
__device__ __forceinline__ float blockReduceMax(float v, float* red) {
    int tid = threadIdx.x;
    red[tid] = v; __syncthreads();
    for (int s = 128; s > 0; s >>= 1) {
        if (tid < s) red[tid] = fmaxf(red[tid], red[tid + s]);
        __syncthreads();
    }
    float r = red[0]; __syncthreads();
    return r;
}

// K1: per-pixel softmax statistics (max over classes, sum of exp)
__global__ void pixel_stats_kernel(const float* __restrict__ seg,
                                   float* __restrict__ pixmax,
                                   float* __restrict__ pixsum) {
    int idx = blockIdx.x * blockDim.x + threadIdx.x;   // b*NP + n
    int b = idx >> 10;
    int n = idx & (NP - 1);
    const float* base = seg + (size_t)b * NC * NP + n;
    float m = -3.4e38f;
    for (int c = 0; c < NC; ++c) m = fmaxf(m, base[(size_t)c * NP]);
    float s = 0.f;
    for (int c = 0; c < NC; ++c) s += __expf(base[(size_t)c * NP] - m);
    pixmax[idx] = m;
    pixsum[idx] = s;
}

// K2: mean probability per (batch, class)
__global__ void class_mean_kernel(const float* __restrict__ seg,
                                  const float* __restrict__ pixmax,
                                  const float* __restrict__ pixsum,
                                  float* __restrict__ meanp) {
    __shared__ float red[256];
    int blk = blockIdx.x;                 // b*NC + c
    int b = blk / NC, c = blk % NC;
    const float* lg = seg + (size_t)b * NC * NP + (size_t)c * NP;
    const float* pm = pixmax + (size_t)b * NP;
    const float* ps = pixsum + (size_t)b * NP;
    float s = 0.f;
    for (int i = 0; i < 4; ++i) {
        int n = threadIdx.x + i * 256;
        s += __expf(lg[n] - pm[n]) / ps[n];
    }
    s = blockReduceSum(s, red);
    if (threadIdx.x == 0) meanp[blk] = s * (1.0f / (float)NP);
}

// K3: iterative top-k (k=11) per batch; strict '>' keeps lowest index on ties (jax.lax.top_k)
__global__ void topk_kernel(const float* __restrict__ meanp,
                            int* __restrict__ selidx,
                            float* __restrict__ out_sel) {
    int b = blockIdx.x * blockDim.x + threadIdx.x;
    if (b >= BATCH) return;
    float vals[NC];
    for (int c = 0; c < NC; ++c) vals[c] = meanp[b * NC + c];
    for (int s = 0; s < NSLOT; ++s) {
        int bi = 0; float bv = -3.4e38f;
        for (int c = 0; c < NC; ++c) {
            if (vals[c] > bv) { bv = vals[c]; bi = c; }
        }
        selidx[b * NSLOT + s] = bi;
        out_sel[b * NSLOT + s] = (float)bi;
        vals[bi] = -3.4e38f;
    }
}

// K4: gather w for selected classes, softmax over pixels -> alpha rows of Wt;
//     emit presence / centroids / areas.
__global__ void row_build_kernel(const float* __restrict__ seg,
                                 const float* __restrict__ pixmax,
                                 const float* __restrict__ pixsum,
                                 const int* __restrict__ selidx,
                                 const float* __restrict__ coordx,
                                 const float* __restrict__ coordy,
                                 float* __restrict__ Wt,
                                 float* __restrict__ out_presence,
                                 float* __restrict__ out_centroids,
                                 float* __restrict__ out_areas) {
    __shared__ float red[256];
    int blk = blockIdx.x;                       // b*NSLOT + s
    int b = blk / NSLOT, s = blk % NSLOT;
    int c = selidx[b * NSLOT + s];
    int tid = threadIdx.x;
    const float* lg = seg + (size_t)b * NC * NP + (size_t)c * NP;
    const float* pm = pixmax + (size_t)b * NP;
    const float* ps = pixsum + (size_t)b * NP;

    float w[4];
    float sw = 0.f, swx = 0.f, swy = 0.f, mw = -3.4e38f;
    for (int i = 0; i < 4; ++i) {
        int n = tid + i * 256;
        w[i] = __expf(lg[n] - pm[n]) / ps[n];
        sw += w[i];
        swx += w[i] * coordx[n];
        swy += w[i] * coordy[n];
        mw = fmaxf(mw, w[i]);
    }
    sw  = blockReduceSum(sw, red);
    swx = blockReduceSum(swx, red);
    swy = blockReduceSum(swy, red);
    mw  = blockReduceMax(mw, red);

    float e[4]; float se = 0.f;
    for (int i = 0; i < 4; ++i) { e[i] = __expf(w[i] - mw); se += e[i]; }
    se = blockReduceSum(se, red);
    float inv = 1.0f / se;

    float* row = Wt + ((size_t)b * NTOK + 1 + s) * NP;
    for (int i = 0; i < 4; ++i) row[tid + i * 256] = e[i] * inv;

    if (tid == 0) {
        float presence = sw * (1.0f / (float)NP);
        float total = sw + 1e-6f;
        out_presence[b * NSLOT + s] = presence;
        out_areas[b * NSLOT + s] = presence;
        out_centroids[(b * NSLOT + s) * 2 + 0] = swx / total;
        out_centroids[(b * NSLOT + s) * 2 + 1] = swy / total;
    }
}

// K5: row 0 = global-mean weights (1/1024); rows 12..15 = region weights (mask/256)
__global__ void fixed_rows_kernel(const float* __restrict__ rmask,
                                  float* __restrict__ Wt) {
    int idx = blockIdx.x * blockDim.x + threadIdx.x;    // b*NP + n
    int b = idx >> 10;
    int n = idx & (NP - 1);
    float* base = Wt + (size_t)b * NTOK * NP;
    base[n] = 1.0f / (float)NP;
    for (int r = 0; r < 4; ++r)
        base[(size_t)(12 + r) * NP + n] = rmask[(size_t)r * NP + n] * (1.0f / 256.0f);
}

// K6: tokens[b] (16x1024) = Wt[b] (16x1024) @ V[b] (1024x1024) via V_WMMA_F32_16X16X4_F32.
// Grid = BATCH*SPLIT blocks (8 waves each); each block owns 256 output columns of one
// batch; each wave owns 2 accumulator tiles (32 columns). The 16x1024 weight strip is
// staged into LDS via the gfx1250 async global->LDS path (ASYNCcnt-tracked).
__global__ __launch_bounds__(256) void gemm_tokens_kernel(const float* __restrict__ V,
                                                          const float* __restrict__ Wt,
                                                          float* __restrict__ out_tokens) {
    __shared__ float As[NTOK * NP];        // 64 KB weight strip
    const int b     = blockIdx.x / SPLIT;
    const int split = blockIdx.x % SPLIT;
    const float* wsrc = Wt + (size_t)b * NTOK * NP;

#if defined(__gfx1250__)
    // Async copy: 16384 floats, 16B per lane per op. VDST = LDS byte offset
    // (low 32 bits of the generic shared pointer), VADDR = 64-bit global address.
    for (int i = threadIdx.x * 4; i < NTOK * NP; i += 256 * 4) {
        uint32_t lds_off = (uint32_t)(uintptr_t)(&As[i]);
        uint64_t gaddr   = (uint64_t)(uintptr_t)(wsrc + i);
        asm volatile("global_load_async_to_lds_b128 %0, %1, off"
                     :: "v"(lds_off), "v"(gaddr) : "memory");
    }
    asm volatile("s_wait_asynccnt 0x0" ::: "memory");
#else
    for (int i = threadIdx.x; i < NTOK * NP; i += 256) As[i] = wsrc[i];
#endif
    __syncthreads();

    const int lane  = threadIdx.x & 31;
    const int wave  = threadIdx.x >> 5;
    const int mrow  = lane & 15;          // M for A / N for B (per ISA VGPR layout)
    const int khalf = lane >> 4;          // K pair select

    const float* vb = V + (size_t)b * NP * HID;
    const int colBase = split * 256 + wave * (TPW * 16);

    v8f acc[TPW] = {};

    for (int k0 = 0; k0 < NP; k0 += 4) {
        // A fragment: lane L holds A[m][k0 + 2*khalf + {0,1}]  (8B aligned -> ds_load_b64)
        const float2 av = *(const float2*)&As[mrow * NP + k0 + khalf * 2];
        v2f a; a.x = av.x; a.y = av.y;

        const float* vr0 = vb + (size_t)(k0 + khalf * 2) * HID;
        #pragma unroll
        for (int t = 0; t < TPW; ++t) {
            int col = colBase + t * 16 + mrow;
            v2f bf;
            bf.x = vr0[col];               // B[k][n]
            bf.y = vr0[HID + col];         // B[k+1][n]
            acc[t] = __builtin_amdgcn_wmma_f32_16x16x4_f32(
                false, a, false, bf, (short)0, acc[t], false, false);
        }
    }

    float* ob = out_tokens + (size_t)b * NTOK * HID;
    #pragma unroll
    for (int t = 0; t < TPW; ++t) {
        int col = colBase + t * 16 + mrow;
        #pragma unroll
        for (int v = 0; v < 8; ++v) {
            int m = v + khalf * 8;         // C/D layout: VGPR v -> M = v + 8*(lane/16)
            ob[(size_t)m * HID + col] = acc[t][v];
        }
    }
}

// ---------------- host launch ----------------
extern "C" void kernel_launch(void* const* d_in, const int* in_sizes, int n_in,
                              void* d_out, int out_size, void* d_ws, size_t ws_size,
                              hipStream_t stream) {
    const float* vision = (const float*)d_in[0];   // (B, 1024, 1024)
    const float* seg    = (const float*)d_in[1];   // (B, 150, 32, 32)
    const float* cx     = (const float*)d_in[2];   // (1024)
    const float* cy     = (const float*)d_in[3];   // (1024)
    const float* rmask  = (const float*)d_in[4];   // (4, 1024)
    // d_in[5] = null_tokens, unused (16 tokens exactly fill MAX_TOKENS)

    float* out = (float*)d_out;
    // output layout (flat, return order): tokens, sel_idx, presence, centroids, areas
    float* out_tokens    = out;                                    // B*16*1024
    float* out_sel       = out + (size_t)BATCH * NTOK * HID;       // B*11
    float* out_presence  = out_sel + BATCH * NSLOT;                // B*11
    float* out_centroids = out_presence + BATCH * NSLOT;           // B*11*2
    float* out_areas     = out_centroids + BATCH * NSLOT * 2;      // B*11

    // workspace layout
    float* wsf    = (float*)d_ws;
    float* pixmax = wsf;                                   // B*NP
    float* pixsum = pixmax + BATCH * NP;                   // B*NP
    float* meanp  = pixsum + BATCH * NP;                   // B*NC
    float* Wt     = meanp + BATCH * NC;                    // B*16*NP
    int*   selidx = (int*)(Wt + (size_t)BATCH * NTOK * NP);// B*11

    // K1: per-pixel softmax stats
    pixel_stats_kernel<<<(BATCH * NP) / 256, 256, 0, stream>>>(seg, pixmax, pixsum);
    // K2: per-class mean probabilities
    class_mean_kernel<<<BATCH * NC, 256, 0, stream>>>(seg, pixmax, pixsum, meanp);
    // K3: top-k selection
    topk_kernel<<<2, 32, 0, stream>>>(meanp, selidx, out_sel);
    // K5: fixed weight rows (independent of K3/K4)
    fixed_rows_kernel<<<(BATCH * NP) / 256, 256, 0, stream>>>(rmask, Wt);
    // K4: alpha rows + presence/centroids/areas
    row_build_kernel<<<BATCH * NSLOT, 256, 0, stream>>>(seg, pixmax, pixsum, selidx,
                                                        cx, cy, Wt,
                                                        out_presence, out_centroids, out_areas);
    // K6: WMMA GEMM -> tokens (split over columns for occupancy)
    gemm_tokens_kernel<<<BATCH * SPLIT, 256, 0, stream>>>(vision, Wt, out_tokens);
}